// CTCLoss_55688545960338
// MI455X (gfx1250) — compile-verified
//
#include <hip/hip_runtime.h>
#include <hip/hip_bf16.h>
#include <stdint.h>

#define CTC_NEG  (-1e30f)
#define CTC_T    256
#define CTC_C    1024
#define CTC_S    32
#define CTC_E    65            // 2*S+1
#define ROWS_PER_BLK 16        // rows of pred handled per (single-wave) block in lse kernel

typedef __attribute__((ext_vector_type(2)))  float        v2f;
typedef __attribute__((ext_vector_type(8)))  float        v8f;
typedef __attribute__((ext_vector_type(4)))  unsigned int v4u;
typedef __attribute__((ext_vector_type(8)))  int          v8i;
typedef __attribute__((ext_vector_type(4)))  int          v4i;

#if __has_builtin(__builtin_amdgcn_tensor_load_to_lds) && __has_builtin(__builtin_amdgcn_s_wait_tensorcnt)
#define CTC_USE_TDM 1
#endif

#if __has_builtin(__builtin_amdgcn_wmma_f32_16x16x4_f32)
#define CTC_USE_WMMA 1
#endif

// ---------------------------------------------------------------------------
// Kernel 1: per-row logsumexp over C=1024.
// One wave (32 threads) per block, 16 rows per block staged in LDS via TDM,
// row-sum of exp() computed with V_WMMA_F32_16X16X4_F32 (B = ones).
// ---------------------------------------------------------------------------
__global__ __launch_bounds__(32)
void ctc_lse_kernel(const float* __restrict__ pred, float* __restrict__ lse)
{
    __shared__ float tile[ROWS_PER_BLK * CTC_C];   // 64 KB
    __shared__ float rowmax[ROWS_PER_BLK];

    const int lane    = threadIdx.x;               // 0..31 (wave32)
    const int rowbase = blockIdx.x * ROWS_PER_BLK;
    const float* gsrc = pred + (size_t)rowbase * CTC_C;

#ifdef CTC_USE_TDM
    {
        // Tensor DMA: 2D tile, 1024 x 16 elements of 4 bytes, contiguous rows.
        const uint64_t gaddr   = (uint64_t)(uintptr_t)gsrc;
        const uint32_t lds_off = (uint32_t)(uintptr_t)(&tile[0]);

        v4u g0;
        g0[0] = 1u;                                         // count=1, user mode
        g0[1] = lds_off;                                    // lds_addr (bytes)
        g0[2] = (uint32_t)(gaddr & 0xFFFFFFFFu);            // global_addr[31:0]
        g0[3] = (uint32_t)((gaddr >> 32) & 0x1FFFFFFu)      // global_addr[56:32]
              | (2u << 30);                                 // type = 2 ("image")

        v8i g1;
        g1[0] = (int)(2u << 16);                            // wg_mask=0, data_size=2 (4B)
        g1[1] = (int)((CTC_C & 0xFFFFu) << 16);             // tensor_dim0[15:0] in [31:16]
        g1[2] = (int)(((unsigned)CTC_C >> 16)               // tensor_dim0[31:16]
              | ((unsigned)ROWS_PER_BLK << 16));            // tensor_dim1[15:0]
        g1[3] = (int)(((unsigned)CTC_C) << 16);             // tensor_dim1 hi=0 | tile_dim0=1024
        g1[4] = (int)ROWS_PER_BLK;                          // tile_dim1=16, tile_dim2=0
        g1[5] = (int)CTC_C;                                 // tensor_dim0_stride[31:0]
        g1[6] = 0;                                          // stride0 hi / stride1 lo
        g1[7] = 0;

        v4i gz = {};
#if defined(__clang_major__) && (__clang_major__ >= 23)
        v8i gz8 = {};
        __builtin_amdgcn_tensor_load_to_lds(g0, g1, gz, gz, gz8, 0);
#else
        __builtin_amdgcn_tensor_load_to_lds(g0, g1, gz, gz, 0);
#endif
        __builtin_amdgcn_s_wait_tensorcnt(0);
    }
#else
    for (int i = lane; i < ROWS_PER_BLK * CTC_C; i += 32)
        tile[i] = gsrc[i];
#endif
    __syncthreads();   // single-wave WG: cheap; orders LDS writes vs. reads

    // Pass 1: per-row max (VALU + wave32 shuffles, conflict-free stride-32 reads)
    for (int r = 0; r < ROWS_PER_BLK; ++r) {
        float m = CTC_NEG;
        const float* rp = &tile[r * CTC_C];
        for (int j = lane; j < CTC_C; j += 32) m = fmaxf(m, rp[j]);
        for (int off = 16; off; off >>= 1)
            m = fmaxf(m, __shfl_xor(m, off, 32));
        if (lane == 0) rowmax[r] = m;
    }
    __syncthreads();

    const int   row  = lane & 15;
    const float rm   = rowmax[row];

#ifdef CTC_USE_WMMA
    // Pass 2: row-sum of exp(x-max) via chained WMMA accumulate, B = ones(4x16).
    // A (16x4 f32): lanes 0-15 hold {K0,K1}, lanes 16-31 hold {K2,K3}.
    const int koff = (lane >> 4) << 1;            // 0 or 2
    v2f bones; bones.x = 1.0f; bones.y = 1.0f;
    v8f acc = {};
    const float* rp = &tile[row * CTC_C];
    for (int k = 0; k < CTC_C; k += 4) {
        v2f a;
        a.x = __expf(rp[k + koff    ] - rm);
        a.y = __expf(rp[k + koff + 1] - rm);
        acc = __builtin_amdgcn_wmma_f32_16x16x4_f32(
                  false, a, false, bones, (short)0, acc, false, false);
    }
    // D layout: VGPR j -> M=j (lanes 0-15) / M=8+j (lanes 16-31), any N column.
    if ((lane < 8) || (lane >= 16 && lane < 24)) {
        const int   r_out = (lane < 8) ? lane : (lane - 8);
        const float s     = acc[lane & 7];
        lse[rowbase + r_out] = rowmax[r_out] + __logf(s);
    }
#else
    // Fallback: shuffle-reduced scalar sums.
    for (int r = 0; r < ROWS_PER_BLK; ++r) {
        float s = 0.0f;
        const float* rp = &tile[r * CTC_C];
        const float mr = rowmax[r];
        for (int j = lane; j < CTC_C; j += 32) s += __expf(rp[j] - mr);
        for (int off = 16; off; off >>= 1) s += __shfl_xor(s, off, 32);
        if (lane == 0) lse[rowbase + r] = mr + __logf(s);
    }
    (void)row; (void)rm;
#endif
}

// ---------------------------------------------------------------------------
// Kernel 2: CTC forward (alpha) recursion. One block per batch sample,
// alpha[E] lives in LDS, T sequential steps with split read/write barriers.
// ---------------------------------------------------------------------------
__global__ __launch_bounds__(96)
void ctc_alpha_kernel(const float* __restrict__ pred,
                      const int*   __restrict__ target,
                      const int*   __restrict__ length,
                      const float* __restrict__ lse,
                      float*       __restrict__ loss_out)
{
    const int b = blockIdx.x;
    const int e = threadIdx.x;

    __shared__ float sa[CTC_E];

    int  label = 0;
    bool skip  = false;
    if (e < CTC_E && (e & 1)) {
        label = target[b * CTC_S + (e >> 1)];
        const int prev2 = (e >= 3) ? target[b * CTC_S + (e >> 1) - 1] : 0;
        skip = (label != prev2);     // label >= 1, so label != blank always holds
    }

    const float* pb = pred + (size_t)b * CTC_T * CTC_C;
    const float* lb = lse  + (size_t)b * CTC_T;

    if (e < CTC_E)
        sa[e] = (e < 2) ? (pb[label] - lb[0]) : CTC_NEG;
    __syncthreads();

    for (int t = 1; t < CTC_T; ++t) {
        float a0 = CTC_NEG, a1 = CTC_NEG, a2 = CTC_NEG, em = 0.0f;
        if (e < CTC_E) {
            a0 = sa[e];
            a1 = (e >= 1) ? sa[e - 1] : CTC_NEG;
            a2 = (skip && e >= 2) ? sa[e - 2] : CTC_NEG;
            em = pb[(size_t)t * CTC_C + label] - lb[t];
        }
        __syncthreads();
        if (e < CTC_E) {
            const float m = fmaxf(a0, fmaxf(a1, a2));
            const float s = __expf(a0 - m) + __expf(a1 - m) + __expf(a2 - m);
            sa[e] = m + __logf(s) + em;
        }
        __syncthreads();
    }

    if (e == 0) {
        const int   L  = length[b];
        const float l1 = sa[2 * L];
        const float l2 = sa[2 * L - 1];
        const float m  = fmaxf(l1, l2);
        const float ll = -(m + __logf(__expf(l1 - m) + __expf(l2 - m)));
        loss_out[b] = ll / (float)L;
    }
}

// ---------------------------------------------------------------------------
// Kernel 3: mean over B per-sample losses.
// ---------------------------------------------------------------------------
__global__ void ctc_mean_kernel(const float* __restrict__ loss_in,
                                float* __restrict__ out, int B)
{
    __shared__ float s[256];
    const int t = threadIdx.x;
    s[t] = (t < B) ? loss_in[t] : 0.0f;
    __syncthreads();
    for (int off = 128; off; off >>= 1) {
        if (t < off) s[t] += s[t + off];
        __syncthreads();
    }
    if (t == 0) out[0] = s[0] / (float)B;
}

// ---------------------------------------------------------------------------
extern "C" void kernel_launch(void* const* d_in, const int* in_sizes, int n_in,
                              void* d_out, int out_size, void* d_ws, size_t ws_size,
                              hipStream_t stream)
{
    (void)n_in; (void)out_size; (void)ws_size;

    const float* pred   = (const float*)d_in[0];
    const int*   target = (const int*)d_in[1];
    const int*   length = (const int*)d_in[2];
    float*       out    = (float*)d_out;

    const int B = in_sizes[2];                 // 128
    const int nrows = B * CTC_T;               // 32768 logsumexp rows

    float* lse    = (float*)d_ws;              // nrows floats
    float* loss_b = lse + nrows;               // B floats

    ctc_lse_kernel<<<nrows / ROWS_PER_BLK, 32, 0, stream>>>(pred, lse);
    ctc_alpha_kernel<<<B, 96, 0, stream>>>(pred, target, length, lse, loss_b);
    ctc_mean_kernel<<<1, 256, 0, stream>>>(loss_b, out, B);
}